// CRFTagger_20641612824889
// MI455X (gfx1250) — compile-verified
//
#include <hip/hip_runtime.h>
#include <hip/hip_bf16.h>
#include <stdint.h>

// CRF tagger: fused forward(logsumexp via WMMA f32 16x16x4) + Viterbi + gold score,
// then backtrace + deterministic NLL reduction.
// L=512, B=4096, T=25 (fixed by the reference).

#define L_ 512
#define B_ 4096
#define T_ 25
#define HSTRIDE 28         // padded backpointer row (bytes), 4B aligned
#define NW 2               // waves per block
#define THREADS (NW * 32)
#define NBLK (NW * 16)     // batch rows per block
#define KPAD 28            // T padded to multiple of 4 for 16x16x4 WMMA
#define RS 28              // alpha/score row stride (floats): 16B-aligned rows,
                           // 28*m mod 64 distinct for m<16 -> conflict-free

typedef __attribute__((ext_vector_type(2))) float v2f;
typedef __attribute__((ext_vector_type(8))) float v8f;
typedef int v4i __attribute__((vector_size(16)));  // matches async-to-LDS builtin pointee

#if defined(__HIP_DEVICE_COMPILE__) && __has_builtin(__builtin_amdgcn_wmma_f32_16x16x4_f32)
#define HAVE_WMMA 1
#else
#define HAVE_WMMA 0
#endif

#if defined(__HIP_DEVICE_COMPILE__) && __has_builtin(__builtin_amdgcn_global_load_async_to_lds_b128)
#define HAVE_ASYNC 1
#else
#define HAVE_ASYNC 0
#endif

// Compiler-only fence: forbids reordering of LDS/memory ops across it (cross-lane
// communication through LDS within one wave; hardware keeps DS ops in order).
__device__ __forceinline__ void wavefence() {
#if defined(__HIP_DEVICE_COMPILE__)
  __builtin_amdgcn_wave_barrier();
#endif
  asm volatile("" ::: "memory");
}

__device__ __forceinline__ void wait_async() {
#if defined(__HIP_DEVICE_COMPILE__)
#if __has_builtin(__builtin_amdgcn_s_wait_asynccnt)
  __builtin_amdgcn_s_wait_asynccnt(0);
#elif HAVE_ASYNC
  asm volatile("s_wait_asynccnt 0" ::: "memory");
#endif
#endif
}

// Raw v_log_f32 (log2) * ln2: inputs here are always normal (>= ~0.6), so we can
// skip __logf's subnormal fixup branch in the hot loop.
__device__ __forceinline__ float fast_log(float x) {
#if defined(__HIP_DEVICE_COMPILE__) && __has_builtin(__builtin_amdgcn_logf)
  return 0.6931471805599453f * __builtin_amdgcn_logf(x);
#else
  return __logf(x);
#endif
}

// Stage one wave's 16x25 f32 emission slice (100 x 16B chunks) into LDS.
__device__ __forceinline__ void stage16(const float* __restrict__ gsrc,
                                        float* ldst, int lane) {
#if HAVE_ASYNC
  for (int c = lane; c < 100; c += 32) {
    __builtin_amdgcn_global_load_async_to_lds_b128(
        (__attribute__((address_space(1))) v4i*)(gsrc + 4 * c),
        (__attribute__((address_space(3))) v4i*)(ldst + 4 * c),
        0, 0);
  }
#else
  for (int c = lane; c < 100; c += 32) {
    float4 v = ((const float4*)gsrc)[c];
    ((float4*)ldst)[c] = v;
  }
#endif
}

// Load 25 consecutive floats (16B-aligned base) into registers.
__device__ __forceinline__ void load25(const float* p, float* r) {
#pragma unroll
  for (int q = 0; q < 6; ++q) {
    float4 v4 = ((const float4*)p)[q];
    r[4 * q + 0] = v4.x;
    r[4 * q + 1] = v4.y;
    r[4 * q + 2] = v4.z;
    r[4 * q + 3] = v4.w;
  }
  r[24] = p[24];
}

__global__ __launch_bounds__(THREADS) void crf_forward_kernel(
    const float* __restrict__ em,     // [L,B,T]
    const int* __restrict__ tags,     // [L,B]
    const float* __restrict__ trans,  // [T,T]
    const float* __restrict__ startT, // [T]
    const float* __restrict__ endT,   // [T]
    uint8_t* __restrict__ hist,       // [L-1,B,HSTRIDE]
    float* __restrict__ nb,           // [B]  numerator - log_z
    int* __restrict__ lastTag)        // [B]
{
  __shared__ float sE[KPAD * 32];                    // exp(trans) padded, 0-filled
  __shared__ __align__(16) float sTrT[T_ * RS];      // transposed trans: [j][i], rows 16B-aligned
  __shared__ float sStart[32];
  __shared__ float sEnd[32];
  __shared__ __align__(16) float sAlpha[NW * 16 * RS];
  __shared__ __align__(16) float sScore[NW * 16 * RS];
  __shared__ float sMrow[NW * 16];
  __shared__ __align__(16) float sEm[2][NBLK * T_];

  const int tid  = threadIdx.x;
  const int wave = tid >> 5;
  const int lane = tid & 31;
  const int mr   = lane & 15;          // matrix row owned within the wave tile
  const int hi   = (lane >= 16) ? 1 : 0;
  const int kofs = hi ? 2 : 0;         // A/B 16x4 f32 lane layout: K offset
  const int b0   = (int)blockIdx.x * NBLK;
  const int b    = b0 + wave * 16 + mr;

  // ---- constant tables to LDS ----
  for (int idx = tid; idx < KPAD * 32; idx += THREADS) {
    int i = idx >> 5, j = idx & 31;
    bool valid = (i < T_) && (j < T_);
    float tv = valid ? trans[i * T_ + j] : 0.f;
    sE[idx] = valid ? __expf(tv) : 0.f;
  }
  for (int idx = tid; idx < T_ * RS; idx += THREADS) {
    int j = idx / RS, i = idx % RS;
    sTrT[idx] = (i < T_) ? trans[i * T_ + j] : 0.f;  // column-major (i padded)
  }
  if (tid < 32) {
    sStart[tid] = (tid < T_) ? startT[tid] : 0.f;
    sEnd[tid]   = (tid < T_) ? endT[tid]   : 0.f;
  }

  // stage t=0 and t=1 emission slices (per-wave, double buffered)
  stage16(em + ((size_t)0 * B_ + b0 + wave * 16) * T_, &sEm[0][wave * 16 * T_], lane);
  stage16(em + ((size_t)1 * B_ + b0 + wave * 16) * T_, &sEm[1][wave * 16 * T_], lane);
  __syncthreads();   // constants visible to both waves
  wait_async();
  wavefence();

  // ---- loop-invariant B fragments: E chunks, 2 N-tiles x 7 K-chunks ----
  v2f bf[2][7];
#pragma unroll
  for (int n = 0; n < 2; ++n)
#pragma unroll
    for (int c = 0; c < 7; ++c)
#pragma unroll
      for (int v = 0; v < 2; ++v)
        bf[n][c][v] = sE[(4 * c + v + kofs) * 32 + (16 * n + mr)];

  float* myA = &sAlpha[wave * 16 * RS];
  float* myS = &sScore[wave * 16 * RS];
  float* myM = &sMrow[wave * 16];

  // ---- init (t = 0) ----
  {
    const float* emt = &sEm[0][wave * 16 * T_];
#pragma unroll 2
    for (int jj = 0; jj < 13; ++jj) {
      int j = hi ? 12 + jj : jj;
      if (hi || jj < 12) {
        float v = sStart[j] + emt[mr * T_ + j];
        myA[mr * RS + j] = v;
        myS[mr * RS + j] = v;
      }
    }
  }
  float num = 0.f;
  int prev = 0;
  if (!hi) {
    int tg = tags[b];
    prev = tg;
    num = sStart[tg] + sEm[0][(wave * 16 + mr) * T_ + tg];
  }
  wavefence();

  // ================= main recurrence =================
  for (int t = 1; t < L_; ++t) {
    const float* emt = &sEm[t & 1][wave * 16 * T_];
    if (t + 1 < L_)
      stage16(em + ((size_t)(t + 1) * B_ + b0 + wave * 16) * T_,
              &sEm[(t + 1) & 1][wave * 16 * T_], lane);

    // ---------- forward: alpha' = log(expA @ E) + m + em ----------
    float arow[28];
    load25(&myA[mr * RS], arow);
    // pad with very-negative so exp(pad - pm) == 0 -> no masking needed
    arow[25] = arow[26] = arow[27] = -1e30f;
    float pm = arow[0];
#pragma unroll
    for (int i = 1; i < T_; ++i) pm = fmaxf(pm, arow[i]);
    if (!hi) myM[mr] = pm;  // both half-lanes computed the identical value

    // A fragment: lane needs arow[4c+v] (lo half) or arow[4c+v+2] (hi half);
    // both are COMPILE-TIME indices -> one cndmask each, no select ladders.
    v2f a[7];
#pragma unroll
    for (int c = 0; c < 7; ++c)
#pragma unroll
      for (int v = 0; v < 2; ++v) {
        float xlo = arow[4 * c + v];
        float xhi = arow[4 * c + v + 2];
        float x = hi ? xhi : xlo;
        a[c][v] = __expf(x - pm);
      }
    wavefence();  // myM store + alpha row reads done before consumption below

    float newA[2][8];
#pragma unroll
    for (int n = 0; n < 2; ++n) {
      float accf[8];
#if HAVE_WMMA
      v8f acc = {0.f, 0.f, 0.f, 0.f, 0.f, 0.f, 0.f, 0.f};
#pragma unroll
      for (int c = 0; c < 7; ++c)
        acc = __builtin_amdgcn_wmma_f32_16x16x4_f32(false, a[c], false, bf[n][c],
                                                    (short)0, acc, false, false);
#pragma unroll
      for (int r = 0; r < 8; ++r) accf[r] = acc[r];
#else
      int jcs = 16 * n + mr;
#pragma unroll
      for (int r = 0; r < 8; ++r) {
        int m2 = r + (hi ? 8 : 0);
        float s = 0.f;
        for (int k = 0; k < T_; ++k)
          s += __expf(myA[m2 * RS + k] - myM[m2]) * sE[k * 32 + jcs];
        accf[r] = s;
      }
#endif
      int jc = 16 * n + mr;  // D layout: lane holds column jc, rows r(+8 for hi)
      if (jc < T_) {
#pragma unroll
        for (int r = 0; r < 8; ++r) {
          int m2 = r + (hi ? 8 : 0);
          newA[n][r] = fast_log(accf[r]) + myM[m2] + emt[m2 * T_ + jc];
        }
      }
    }
    wavefence();  // all reads of old alpha complete before writeback
#pragma unroll
    for (int n = 0; n < 2; ++n) {
      int jc = 16 * n + mr;
      if (jc < T_)
#pragma unroll
        for (int r = 0; r < 8; ++r)
          myA[(r + (hi ? 8 : 0)) * RS + jc] = newA[n][r];
    }

    // ---------- Viterbi (max-plus), array-free / low register pressure ----------
    float srow[25];
    load25(&myS[mr * RS], srow);
    wavefence();  // all score-row reads done before any lane writes new scores

    uint8_t* hb = hist + ((size_t)(t - 1) * B_ + b) * HSTRIDE;
    const int off0 = hi ? 12 : 0;
    uint32_t pk = 0;
#pragma unroll 2
    for (int jj = 0; jj < 13; ++jj) {
      int j = hi ? 12 + jj : jj;
      bool valid = hi || (jj < 12);
      float bv = -1e30f;
      int bi = 0;
      if (valid) {
        float trc[25];
        load25(&sTrT[j * RS], trc);  // contiguous transposed column
#pragma unroll
        for (int i = 0; i < T_; ++i) {
          float v = srow[i] + trc[i];
          if (v > bv) { bv = v; bi = i; }  // strict > keeps first max (argmax semantics)
        }
        myS[mr * RS + j] = bv + emt[mr * T_ + j];
      }
      pk = (pk >> 8) | ((uint32_t)bi << 24);
      if ((jj & 3) == 3) *(uint32_t*)(hb + off0 + (jj >> 2) * 4) = pk;
      if (hi && jj == 12) hb[24] = (uint8_t)bi;
    }

    // ---------- gold-path score ----------
    if (!hi) {
      int tg = tags[(size_t)t * B_ + b];
      num += sTrT[tg * RS + prev] + emt[mr * T_ + tg];
      prev = tg;
    }

    wait_async();
    wavefence();
  }

  // ---- finalize: log_z and last tag ----
  float pmz = -1e30f;
#pragma unroll
  for (int jj = 0; jj < 13; ++jj) {
    int j = hi ? 12 + jj : jj;
    if (hi || jj < 12) pmz = fmaxf(pmz, myA[mr * RS + j] + sEnd[j]);
  }
  pmz = fmaxf(pmz, __shfl_xor(pmz, 16));
  float ps = 0.f;
#pragma unroll
  for (int jj = 0; jj < 13; ++jj) {
    int j = hi ? 12 + jj : jj;
    if (hi || jj < 12) ps += __expf(myA[mr * RS + j] + sEnd[j] - pmz);
  }
  ps += __shfl_xor(ps, 16);
  float lz = fast_log(ps) + pmz;

  float bv = -1e30f;
  int bj = 0;
#pragma unroll
  for (int jj = 0; jj < 13; ++jj) {
    int j = hi ? 12 + jj : jj;
    if (hi || jj < 12) {
      float v = myS[mr * RS + j] + sEnd[j];
      if (v > bv) { bv = v; bj = j; }
    }
  }
  float ov = __shfl_xor(bv, 16);
  int oj = __shfl_xor(bj, 16);
  if (!hi) {
    if (ov > bv) { bv = ov; bj = oj; }  // hi half holds larger j; strict > -> first max
    num += sEnd[prev];
    nb[b] = num - lz;
    lastTag[b] = bj;
  }
}

__global__ void crf_backtrace(const uint8_t* __restrict__ hist,
                              const int* __restrict__ lastTag,
                              float* __restrict__ outPath) {
  int b = blockIdx.x * blockDim.x + threadIdx.x;
  if (b >= B_) return;
  int cur = lastTag[b];
  outPath[(size_t)(L_ - 1) * B_ + b] = (float)cur;
  for (int t = L_ - 2; t >= 0; --t) {
    cur = hist[((size_t)t * B_ + b) * HSTRIDE + cur];
    outPath[(size_t)t * B_ + b] = (float)cur;
  }
}

__global__ __launch_bounds__(1024) void crf_reduce(const float* __restrict__ nb,
                                                   float* __restrict__ out0) {
  __shared__ float s[1024];
  int tid = threadIdx.x;
  float v = nb[tid] + nb[tid + 1024] + nb[tid + 2048] + nb[tid + 3072];
  s[tid] = v;
  __syncthreads();
  for (int o = 512; o > 0; o >>= 1) {
    if (tid < o) s[tid] += s[tid + o];
    __syncthreads();
  }
  if (tid == 0) out0[0] = -s[0] / (float)B_;
}

extern "C" void kernel_launch(void* const* d_in, const int* in_sizes, int n_in,
                              void* d_out, int out_size, void* d_ws, size_t ws_size,
                              hipStream_t stream) {
  const float* em    = (const float*)d_in[0];
  const int*   tags  = (const int*)d_in[1];
  const float* trans = (const float*)d_in[2];
  const float* st    = (const float*)d_in[3];
  const float* en    = (const float*)d_in[4];
  float* out = (float*)d_out;  // [0]=nll, [1..] = best_path [L,B] as float

  // workspace layout: hist u8 [L-1][B][HSTRIDE] | nb f32 [B] | lastTag i32 [B]
  uint8_t* hist = (uint8_t*)d_ws;
  size_t histBytes = (size_t)(L_ - 1) * B_ * HSTRIDE;  // ~58.6 MB
  float* nb = (float*)((char*)d_ws + histBytes);
  int* lastTag = (int*)((char*)d_ws + histBytes + (size_t)B_ * sizeof(float));

  crf_forward_kernel<<<B_ / NBLK, THREADS, 0, stream>>>(em, tags, trans, st, en,
                                                        hist, nb, lastTag);
  crf_backtrace<<<B_ / 256, 256, 0, stream>>>(hist, lastTag, out + 1);
  crf_reduce<<<1, 1024, 0, stream>>>(nb, out);
}